// ExcEmbedding_65214783422654
// MI455X (gfx1250) — compile-verified
//
#include <hip/hip_runtime.h>
#include <math.h>

#define NB   256   // batch (both caption and image batch)
#define NR   36    // regions
#define NT   64    // words
#define ND   1024  // embed dim
#define NSQ  128   // squeeze dim

typedef float v2f __attribute__((ext_vector_type(2)));
typedef float v8f __attribute__((ext_vector_type(8)));

__device__ __forceinline__ float leaky01(float x) { return x > 0.f ? x : 0.1f * x; }

// ---------------------------------------------------------------------------
// Kernel 1: cap_vec[b,d] = (sum_{t<len} leaky(cap[b,t,d])) /
//                          ((sqrt(sum_t leaky^2) + 1e-8) * len)
// One thread per (b,d); loop over t with stride ND -> fully coalesced per block.
// ---------------------------------------------------------------------------
__global__ void k_capvec(const float* __restrict__ cap, const int* __restrict__ lens,
                         float* __restrict__ cap_vec) {
  int tid = blockIdx.x * blockDim.x + threadIdx.x;   // 0 .. NB*ND-1
  int b = tid >> 10;
  int d = tid & (ND - 1);
  int len = lens[b];
  const float* p = cap + (size_t)b * NT * ND + d;
  float sq = 0.f, ms = 0.f;
#pragma unroll 8
  for (int t = 0; t < NT; ++t) {
    float l = leaky01(p[(size_t)t * ND]);
    sq += l * l;
    if (t < len) ms += l;
  }
  cap_vec[tid] = ms / ((sqrtf(sq) + 1e-8f) * (float)len);
}

// ---------------------------------------------------------------------------
// Kernel 2: img_vec[b,d] = (sum_r leaky(img[b,r,d])) /
//                          ((sqrt(sum_r leaky^2) + 1e-8) * R)
// ---------------------------------------------------------------------------
__global__ void k_imgvec(const float* __restrict__ img, float* __restrict__ img_vec) {
  int tid = blockIdx.x * blockDim.x + threadIdx.x;
  int b = tid >> 10;
  int d = tid & (ND - 1);
  const float* p = img + (size_t)b * NR * ND + d;
  float sq = 0.f, ms = 0.f;
#pragma unroll 6
  for (int r = 0; r < NR; ++r) {
    float l = leaky01(p[(size_t)r * ND]);
    sq += l * l;
    ms += l;
  }
  img_vec[tid] = ms / ((sqrtf(sq) + 1e-8f) * (float)NR);
}

// ---------------------------------------------------------------------------
// Kernel 3: SE gate.  One block per caption b (256 threads).
//   hidden = relu(cap_vec @ W_sq + b_sq)        (1024 -> 128)
//   gate   = sigmoid(hidden @ W_ex + b_ex)      (128 -> 1024)
// W_sq is (D, d_sq) row-major -> W_sq[d*128+j] coalesced across j.
// W_ex is (d_sq, D) row-major -> W_ex[j*1024+d] coalesced across d.
// ---------------------------------------------------------------------------
__global__ void k_gate(const float* __restrict__ cap_vec,
                       const float* __restrict__ Wsq, const float* __restrict__ bsq,
                       const float* __restrict__ Wex, const float* __restrict__ bex,
                       float* __restrict__ gate) {
  __shared__ float s_cap[ND];
  __shared__ float s_hid[NSQ];
  int b = blockIdx.x;
  int t = threadIdx.x;  // 0..255
#pragma unroll
  for (int q = 0; q < 4; ++q) s_cap[t + 256 * q] = cap_vec[(size_t)b * ND + t + 256 * q];
  __syncthreads();
  if (t < NSQ) {
    float acc = bsq[t];
    for (int d = 0; d < ND; ++d) acc = fmaf(s_cap[d], Wsq[(size_t)d * NSQ + t], acc);
    s_hid[t] = fmaxf(acc, 0.f);
  }
  __syncthreads();
#pragma unroll
  for (int q = 0; q < 4; ++q) {
    int d = t + 256 * q;
    float acc = bex[d];
#pragma unroll 8
    for (int j = 0; j < NSQ; ++j) acc = fmaf(s_hid[j], Wex[(size_t)j * ND + d], acc);
    gate[(size_t)b * ND + d] = 1.f / (1.f + expf(-acc));
  }
}

// ---------------------------------------------------------------------------
// Kernel 4: cap_v = l2norm(cap_vec over D);  S[b] = sum_d cap_v[b,d]
// One block per b, 256 threads, 4 elements each, tree reductions in LDS.
// ---------------------------------------------------------------------------
__global__ void k_capv(const float* __restrict__ cap_vec, float* __restrict__ cap_v,
                       float* __restrict__ S) {
  __shared__ float red[256];
  int b = blockIdx.x, t = threadIdx.x;
  float x[4];
  float sq = 0.f;
#pragma unroll
  for (int q = 0; q < 4; ++q) {
    x[q] = cap_vec[(size_t)b * ND + t + 256 * q];
    sq += x[q] * x[q];
  }
  red[t] = sq;
  __syncthreads();
  for (int s = 128; s > 0; s >>= 1) {
    if (t < s) red[t] += red[t + s];
    __syncthreads();
  }
  float inv = 1.f / (sqrtf(red[0]) + 1e-8f);
  __syncthreads();
  float ssum = 0.f;
#pragma unroll
  for (int q = 0; q < 4; ++q) {
    float cv = x[q] * inv;
    cap_v[(size_t)b * ND + t + 256 * q] = cv;
    ssum += cv;
  }
  red[t] = ssum;
  __syncthreads();
  for (int s = 128; s > 0; s >>= 1) {
    if (t < s) red[t] += red[t + s];
    __syncthreads();
  }
  if (t == 0) S[b] = red[0];
}

// ---------------------------------------------------------------------------
// Kernel 5: fused triple-GEMM + epilogue via V_WMMA_F32_16X16X4_F32.
//   P = v . g^T, Q = v^2 . (g^2)^T, R = v . (g*cv)^T   (all 256x256, K=1024)
//   sims[i,c] = (R + beta*S[c]) / (sqrt(Q + 2*beta*P + beta^2*D) + 1e-8)
// One wave per 16x16 tile; 8 waves/block, 32 blocks. EXEC is all-1s.
// A layout (16x4 f32): lane l -> M=l&15, float2 at K = 2*(l>>4).
// B layout (4x16 f32): lane l -> N=l&15, float2 at K = 2*(l>>4).
// C/D layout: vgpr r, lane l -> M = r + 8*(l>>4), N = l&15.
// ---------------------------------------------------------------------------
__global__ void k_sims(const float* __restrict__ v,   // img_vectors (Bi, D)
                       const float* __restrict__ g,   // gate        (Bc, D)
                       const float* __restrict__ cv,  // cap_v       (Bc, D)
                       const float* __restrict__ S,   // row sums    (Bc,)
                       const float* __restrict__ beta_p,
                       float* __restrict__ out) {     // sims (Bi, Bc)
  int wave = threadIdx.x >> 5;
  int lane = threadIdx.x & 31;
  int tile = blockIdx.x * 8 + wave;   // 0..255
  int i0 = (tile >> 4) * 16;          // image-tile origin (M)
  int c0 = (tile & 15) * 16;          // caption-tile origin (N)
  int ln = lane & 15;
  int lh = lane >> 4;

  float beta = beta_p[0];
  const float* arow  = v  + (size_t)(i0 + ln) * ND;
  const float* grow  = g  + (size_t)(c0 + ln) * ND;
  const float* cvrow = cv + (size_t)(c0 + ln) * ND;

  v8f accP = {};
  v8f accQ = {};
  v8f accR = {};

  for (int d0 = 0; d0 < ND; d0 += 4) {
    int off = d0 + 2 * lh;
    v2f a  = *(const v2f*)(arow + off);
    v2f bg = *(const v2f*)(grow + off);
    v2f bc = *(const v2f*)(cvrow + off);
    v2f a2  = a * a;
    v2f bg2 = bg * bg;
    v2f bgc = bg * bc;
    accP = __builtin_amdgcn_wmma_f32_16x16x4_f32(false, a,  false, bg,  (short)0, accP, false, false);
    accQ = __builtin_amdgcn_wmma_f32_16x16x4_f32(false, a2, false, bg2, (short)0, accQ, false, false);
    accR = __builtin_amdgcn_wmma_f32_16x16x4_f32(false, a,  false, bgc, (short)0, accR, false, false);
  }

  float b2d = beta * beta * (float)ND;
  int c = c0 + ln;
  float bS = beta * S[c];
#pragma unroll
  for (int r = 0; r < 8; ++r) {
    int i = i0 + r + 8 * lh;
    float denom = sqrtf(accQ[r] + 2.f * beta * accP[r] + b2d) + 1e-8f;
    out[(size_t)i * NB + c] = (accR[r] + bS) / denom;
  }
}

// ---------------------------------------------------------------------------
extern "C" void kernel_launch(void* const* d_in, const int* in_sizes, int n_in,
                              void* d_out, int out_size, void* d_ws, size_t ws_size,
                              hipStream_t stream) {
  const float* img_embed = (const float*)d_in[0];  // (256, 36, 1024)
  const float* cap_embed = (const float*)d_in[1];  // (256, 64, 1024)
  const int*   lens      = (const int*)d_in[2];    // (256,)
  const float* W_sq      = (const float*)d_in[3];  // (1024, 128)
  const float* b_sq      = (const float*)d_in[4];  // (128,)
  const float* W_ex      = (const float*)d_in[5];  // (128, 1024)
  const float* b_ex      = (const float*)d_in[6];  // (1024,)
  const float* beta      = (const float*)d_in[7];  // (1,)
  // d_in[8] = beta1, unused in forward
  float* out = (float*)d_out;                      // (256, 256) = sims[i, c]

  float* ws      = (float*)d_ws;
  float* cap_vec = ws;                       // 256*1024
  float* img_vec = cap_vec + NB * ND;        // 256*1024
  float* gate    = img_vec + NB * ND;        // 256*1024
  float* cap_v   = gate + NB * ND;           // 256*1024
  float* Ssum    = cap_v + NB * ND;          // 256

  k_capvec<<<(NB * ND) / 256, 256, 0, stream>>>(cap_embed, lens, cap_vec);
  k_imgvec<<<(NB * ND) / 256, 256, 0, stream>>>(img_embed, img_vec);
  k_gate  <<<NB, 256, 0, stream>>>(cap_vec, W_sq, b_sq, W_ex, b_ex, gate);
  k_capv  <<<NB, 256, 0, stream>>>(cap_vec, cap_v, Ssum);
  k_sims  <<<32, 256, 0, stream>>>(img_vec, gate, cap_v, Ssum, beta, out);
}